// StateReprModuleU_5592047419689
// MI455X (gfx1250) — compile-verified
//
#include <hip/hip_runtime.h>
#include <hip/hip_bf16.h>
#include <stdint.h>

#define BATCH    4096
#define N_HIST   20
#define EMB      64
#define N_ROW    210              // 20 user_item rows + 190 pair rows
#define OUT_STRIDE (N_ROW * EMB)  // 13440 floats per batch element

typedef __attribute__((ext_vector_type(2))) float v2f;
typedef __attribute__((ext_vector_type(2))) int   v2i;

#define AS3 __attribute__((address_space(3)))
#define AS1 __attribute__((address_space(1)))

typedef AS1 v2i* gbl_v2i_p;   // global-AS pointer to 64-bit chunk
typedef AS3 v2i* lds_v2i_p;   // LDS-AS pointer to 64-bit chunk

#if __has_builtin(__builtin_amdgcn_global_load_async_to_lds_b64)
#define HAVE_ASYNC_BUILTIN 1
#endif

__global__ __launch_bounds__(32) void srm_kernel(
    const int*   __restrict__ user,       // (BATCH, 1)
    const int*   __restrict__ hist,       // (BATCH, N_HIST)
    const float* __restrict__ uet,        // (USER_NUM, EMB)
    const float* __restrict__ iet,        // (ITEM_NUM+1, EMB)
    const float* __restrict__ weights,    // (N_HIST,)
    float*       __restrict__ out)        // (BATCH, N_ROW*EMB)
{
    __shared__ float lds_we[N_HIST * EMB];   // 5 KB staging buffer

    const int b    = blockIdx.x;
    const int lane = threadIdx.x;            // 0..31 (wave32)
    const int d    = lane * 2;               // this lane owns dims d, d+1

    // ---- user embedding (L2-resident table, regular temporal load) ----
    const int u = user[b];
    const v2f ue = *(const v2f*)(uet + (size_t)u * EMB + d);

    // ---- async gather of the 20 item rows into LDS (CDNA5 async path) ----
#pragma unroll
    for (int h = 0; h < N_HIST; ++h) {
        const int idx = hist[b * N_HIST + h];           // wave-uniform
        const float* src = iet + (size_t)idx * EMB + d; // per-lane 8B
        float* dst = &lds_we[h * EMB + d];
#if defined(HAVE_ASYNC_BUILTIN)
        __builtin_amdgcn_global_load_async_to_lds_b64(
            (gbl_v2i_p)src, (lds_v2i_p)dst, /*offset=*/0, /*cpol=*/0);
#else
        uint32_t lds_off = (uint32_t)(uintptr_t)(AS3 float*)dst;
        uint64_t gaddr   = (uint64_t)(uintptr_t)src;
        asm volatile("global_load_async_to_lds_b64 %0, %1, off"
                     :: "v"(lds_off), "v"(gaddr) : "memory");
#endif
    }

#if __has_builtin(__builtin_amdgcn_s_wait_asynccnt)
    __builtin_amdgcn_s_wait_asynccnt(0);
#else
    asm volatile("s_wait_asynccnt 0x0" ::: "memory");
#endif
    asm volatile("" ::: "memory");  // compiler ordering barrier for LDS reads

    // ---- read back + apply weights: we[h] = weights[h] * item_emb[h] ----
    v2f we[N_HIST];
#pragma unroll
    for (int h = 0; h < N_HIST; ++h) {
        const v2f e = *(const v2f*)(&lds_we[h * EMB + d]);
        we[h] = e * weights[h];
    }

    // ---- streaming output: NT stores (220 MB > 192 MB L2, don't pollute) ----
    float* ob = out + (size_t)b * OUT_STRIDE + d;

    // user_item rows: 20 x (ue * we[h])
#pragma unroll
    for (int h = 0; h < N_HIST; ++h) {
        const v2f r = ue * we[h];
        __builtin_nontemporal_store(r, (v2f*)(ob + h * EMB));
    }

    // pair rows: upper triangle (i<j), row-major order == jnp.triu_indices(k=1)
    int p = N_HIST;
#pragma unroll
    for (int i = 0; i < N_HIST - 1; ++i) {
#pragma unroll
        for (int j = i + 1; j < N_HIST; ++j) {
            const v2f r = we[i] * we[j];
            __builtin_nontemporal_store(r, (v2f*)(ob + p * EMB));
            ++p;
        }
    }
}

extern "C" void kernel_launch(void* const* d_in, const int* in_sizes, int n_in,
                              void* d_out, int out_size, void* d_ws, size_t ws_size,
                              hipStream_t stream) {
    const int*   user    = (const int*)d_in[0];
    const int*   hist    = (const int*)d_in[1];
    const float* uet     = (const float*)d_in[2];
    const float* iet     = (const float*)d_in[3];
    const float* weights = (const float*)d_in[4];
    float*       out     = (float*)d_out;

    srm_kernel<<<dim3(BATCH), dim3(32), 0, stream>>>(user, hist, uet, iet, weights, out);
}